// ConvolutionalLayer_5875515261084
// MI455X (gfx1250) — compile-verified
//
#include <hip/hip_runtime.h>
#include <hip/hip_bf16.h>
#include <cstdint>

typedef float v2f __attribute__((ext_vector_type(2)));
typedef float v8f __attribute__((ext_vector_type(8)));

#define SA_STRIDE 130   // A tile rows: 128 cols padded (even => 8B aligned v2f, conflict-free)
#define SW1_STRIDE 130  // W1^T rows: 128 cols padded
#define SW2_STRIDE 66   // W2^T rows: 64 cols padded
#define SY_STRIDE 68    // y1 tile rows: 64 cols padded

__device__ inline v8f wmma_f32_16x16x4(v2f a, v2f b, v8f c) {
    // D(16x16,f32) = A(16x4,f32) x B(4x16,f32) + C
    return __builtin_amdgcn_wmma_f32_16x16x4_f32(false, a, false, b, (short)0, c,
                                                 false, false);
}

// ---------------- K0: zero scratch ----------------
__global__ void k_zero(float* __restrict__ p, long long n) {
    long long i = (long long)blockIdx.x * blockDim.x + threadIdx.x;
    long long stride = (long long)gridDim.x * blockDim.x;
    for (; i < n; i += stride) p[i] = 0.0f;
}

// ---------------- K1: scatter-add nbr_sum[dst] += x[src]; multiplicity counts --------
__global__ __launch_bounds__(256) void k_scatter(
    const float* __restrict__ x, const int* __restrict__ src, const int* __restrict__ dst,
    float* __restrict__ nbr, float* __restrict__ cnt_src, float* __restrict__ deg_dst,
    int Ecnt) {
    long long t = (long long)blockIdx.x * blockDim.x + threadIdx.x;
    long long e = t >> 2;
    int q = (int)(t & 3);                 // 16-channel slice
    if (e >= Ecnt) return;
    int s = src[e], d = dst[e];
    const float4* xr = (const float4*)(x + (size_t)s * 64 + q * 16);
    float* nr = nbr + (size_t)d * 64 + q * 16;
    #pragma unroll
    for (int i = 0; i < 4; ++i) {
        float4 v = xr[i];
        atomicAdd(&nr[i * 4 + 0], v.x);
        atomicAdd(&nr[i * 4 + 1], v.y);
        atomicAdd(&nr[i * 4 + 2], v.z);
        atomicAdd(&nr[i * 4 + 3], v.w);
    }
    if (q == 0) {
        atomicAdd(&cnt_src[s], 1.0f);
        atomicAdd(&deg_dst[d], 1.0f);
    }
}

// ---------------- K2: BN1 stats in O(N): sum_e h[e,c] = sum_n mult[n]*val[n,c] --------
__global__ __launch_bounds__(256) void k_bn1_stats(
    const float* __restrict__ x, const float* __restrict__ nbr,
    const float* __restrict__ cnt_src, const float* __restrict__ deg_dst,
    float* __restrict__ g_sum1, float* __restrict__ g_ss1, int Nn) {
    __shared__ float s[128], ss[128];
    int tid = threadIdx.x;
    if (tid < 128) { s[tid] = 0.f; ss[tid] = 0.f; }
    __syncthreads();
    int c = tid & 63;
    int rofs = tid >> 6;
    float as = 0.f, ass = 0.f, bs = 0.f, bss = 0.f;
    for (int n = blockIdx.x * 4 + rofs; n < Nn; n += gridDim.x * 4) {
        float cs = cnt_src[n], dd = deg_dst[n];
        float vx = x[(size_t)n * 64 + c];
        float vn = nbr[(size_t)n * 64 + c];
        as += cs * vx;  ass += cs * vx * vx;
        bs += dd * vn;  bss += dd * vn * vn;
    }
    atomicAdd(&s[c], as);       atomicAdd(&ss[c], ass);
    atomicAdd(&s[64 + c], bs);  atomicAdd(&ss[64 + c], bss);
    __syncthreads();
    if (tid < 128) { atomicAdd(&g_sum1[tid], s[tid]); atomicAdd(&g_ss1[tid], ss[tid]); }
}

// ---------------- K3: fold BN1 into W1 (single block, 128 thr) ----------------
__global__ void k_fold1(const float* __restrict__ g_sum1, const float* __restrict__ g_ss1,
                        const float* __restrict__ gamma1, const float* __restrict__ beta1,
                        const float* __restrict__ W1, const float* __restrict__ b1,
                        float* __restrict__ w1eff, float* __restrict__ b1eff, float invE) {
    __shared__ float sT[128];
    int k = threadIdx.x;                       // 0..127
    float m = g_sum1[k] * invE;
    float v = g_ss1[k] * invE - m * m;
    float sc = gamma1[k] * rsqrtf(v + 1e-5f);
    sT[k] = beta1[k] - m * sc;
    for (int j = 0; j < 64; ++j) w1eff[k * 64 + j] = sc * W1[k * 64 + j];
    __syncthreads();
    if (k < 64) {
        float acc = b1[k];
        for (int kk = 0; kk < 128; ++kk) acc += sT[kk] * W1[kk * 64 + k];
        b1eff[k] = acc;
    }
}

// ---------------- K5: fold BN2 into W2 (single block, 64 thr) ----------------
__global__ void k_fold2(const float* __restrict__ g_sum2, const float* __restrict__ g_ss2,
                        const float* __restrict__ gamma2, const float* __restrict__ beta2,
                        const float* __restrict__ W2, const float* __restrict__ b2,
                        float* __restrict__ w2eff, float* __restrict__ b2eff, float invE) {
    __shared__ float sT[64];
    int k = threadIdx.x;                       // 0..63
    float m = g_sum2[k] * invE;
    float v = g_ss2[k] * invE - m * m;
    float sc = gamma2[k] * rsqrtf(v + 1e-5f);
    sT[k] = beta2[k] - m * sc;
    for (int j = 0; j < 64; ++j) w2eff[k * 64 + j] = sc * W2[k * 64 + j];
    __syncthreads();
    float acc = b2[k];
    for (int kk = 0; kk < 64; ++kk) acc += sT[kk] * W2[kk * 64 + k];
    b2eff[k] = acc;
}

// Cooperative load of one 16-edge x 128-ch A tile into LDS (128 threads)
__device__ inline void load_tileA(float* sA, const float* __restrict__ x,
                                  const float* __restrict__ nbr,
                                  const int* __restrict__ src, const int* __restrict__ dst,
                                  int t, int Ecnt, int tid) {
    int m = tid >> 3;          // edge-in-tile 0..15
    int p = tid & 7;           // 16-float slice 0..7
    int cbase = p * 16;
    int e = t * 16 + m;
    float* d = &sA[m * SA_STRIDE + cbase];
    if (e < Ecnt) {
        const float4* rowp = (cbase < 64)
            ? (const float4*)(x + (size_t)src[e] * 64 + cbase)
            : (const float4*)(nbr + (size_t)dst[e] * 64 + (cbase - 64));
        #pragma unroll
        for (int q = 0; q < 4; ++q) {
            float4 v = rowp[q];
            d[q * 4 + 0] = v.x; d[q * 4 + 1] = v.y;
            d[q * 4 + 2] = v.z; d[q * 4 + 3] = v.w;
        }
    } else {
        #pragma unroll
        for (int q = 0; q < 16; ++q) d[q] = 0.f;
    }
}

// WMMA chain: one wave computes a 16x16 D tile, K accumulated in steps of 4
template <int NK4>
__device__ inline v8f gemm_frag(const float* sA, int strideA, const float* sWt,
                                int strideW, int lane, int col) {
    int khalf = lane >> 4;
    int mrow = lane & 15;
    v8f acc = {};
    #pragma unroll
    for (int kk = 0; kk < NK4; ++kk) {
        int k0 = kk * 4 + khalf * 2;
        v2f a = *(const v2f*)(sA + mrow * strideA + k0);   // A[m][k0..k0+1]
        v2f b = *(const v2f*)(sWt + col * strideW + k0);   // W^T[n][k0..k0+1]
        acc = wmma_f32_16x16x4(a, b, acc);
    }
    return acc;
}

// ---------------- K4: GEMM1 + ReLU + BN2 statistics (y1 not materialized) ------------
__global__ __launch_bounds__(128) void k_gemm1_stats(
    const float* __restrict__ x, const int* __restrict__ src, const int* __restrict__ dst,
    const float* __restrict__ nbr, const float* __restrict__ w1eff,
    const float* __restrict__ b1eff, float* __restrict__ g_sum2, float* __restrict__ g_ss2,
    int nTiles, int Ecnt) {
    __shared__ __align__(16) float sA[16 * SA_STRIDE];
    __shared__ __align__(16) float sW1[64 * SW1_STRIDE];
    __shared__ float sSum[64], sSS[64];
    int tid = threadIdx.x;
    for (int i = tid; i < 128 * 64; i += 128) {       // W1eff^T: sW1[n][k]
        int k = i >> 6, n = i & 63;
        sW1[n * SW1_STRIDE + k] = w1eff[i];
    }
    if (tid < 64) { sSum[tid] = 0.f; sSS[tid] = 0.f; }
    __syncthreads();
    int lane = tid & 31;
    int wv = tid >> 5;                                 // wave = N-tile
    int col = wv * 16 + (lane & 15);
    float bc1 = b1eff[col];
    for (int t = blockIdx.x; t < nTiles; t += gridDim.x) {
        load_tileA(sA, x, nbr, src, dst, t, Ecnt, tid);
        __syncthreads();
        v8f acc = gemm_frag<32>(sA, SA_STRIDE, sW1, SW1_STRIDE, lane, col);
        float s = 0.f, q = 0.f;
        #pragma unroll
        for (int r = 0; r < 8; ++r) {
            float v = fmaxf(acc[r] + bc1, 0.f);        // y1 element
            s += v; q += v * v;
        }
        s += __shfl_xor(s, 16);                        // merge row-halves sharing a column
        q += __shfl_xor(q, 16);
        if ((lane >> 4) == 0) { atomicAdd(&sSum[col], s); atomicAdd(&sSS[col], q); }
        __syncthreads();
    }
    __syncthreads();
    if (tid < 64) { atomicAdd(&g_sum2[tid], sSum[tid]); atomicAdd(&g_ss2[tid], sSS[tid]); }
}

// ---------------- K6: recompute GEMM1, then GEMM2 + ReLU -> out ----------------
__global__ __launch_bounds__(128) void k_gemm2_out(
    const float* __restrict__ x, const int* __restrict__ src, const int* __restrict__ dst,
    const float* __restrict__ nbr, const float* __restrict__ w1eff,
    const float* __restrict__ b1eff, const float* __restrict__ w2eff,
    const float* __restrict__ b2eff, float* __restrict__ out, int nTiles, int Ecnt) {
    __shared__ __align__(16) float sA[16 * SA_STRIDE];
    __shared__ __align__(16) float sW1[64 * SW1_STRIDE];
    __shared__ __align__(16) float sW2[64 * SW2_STRIDE];
    __shared__ __align__(16) float sY[16 * SY_STRIDE];
    int tid = threadIdx.x;
    for (int i = tid; i < 128 * 64; i += 128) {
        int k = i >> 6, n = i & 63;
        sW1[n * SW1_STRIDE + k] = w1eff[i];
    }
    for (int i = tid; i < 64 * 64; i += 128) {
        int k = i >> 6, n = i & 63;
        sW2[n * SW2_STRIDE + k] = w2eff[i];
    }
    __syncthreads();
    int lane = tid & 31;
    int wv = tid >> 5;
    int col = wv * 16 + (lane & 15);
    int khalf = lane >> 4;
    float bc1 = b1eff[col];
    float bc2 = b2eff[col];
    for (int t = blockIdx.x; t < nTiles; t += gridDim.x) {
        load_tileA(sA, x, nbr, src, dst, t, Ecnt, tid);
        __syncthreads();
        // stage 1: y1 tile (each wave computes 16 of 64 columns)
        v8f acc = gemm_frag<32>(sA, SA_STRIDE, sW1, SW1_STRIDE, lane, col);
        #pragma unroll
        for (int r = 0; r < 8; ++r)
            sY[(r + 8 * khalf) * SY_STRIDE + col] = fmaxf(acc[r] + bc1, 0.f);
        __syncthreads();
        // stage 2: out tile = relu(y1 @ W2eff + b2eff)
        v8f acc2 = gemm_frag<16>(sY, SY_STRIDE, sW2, SW2_STRIDE, lane, col);
        #pragma unroll
        for (int r = 0; r < 8; ++r) {
            int e = t * 16 + r + 8 * khalf;
            if (e < Ecnt) out[(size_t)e * 64 + col] = fmaxf(acc2[r] + bc2, 0.f);
        }
        __syncthreads();
    }
}

extern "C" void kernel_launch(void* const* d_in, const int* in_sizes, int n_in,
                              void* d_out, int out_size, void* d_ws, size_t ws_size,
                              hipStream_t stream) {
    const float* x      = (const float*)d_in[0];
    const int*   src    = (const int*)  d_in[1];
    const int*   dst    = (const int*)  d_in[2];
    const float* gamma1 = (const float*)d_in[3];
    const float* beta1  = (const float*)d_in[4];
    const float* W1     = (const float*)d_in[5];
    const float* b1     = (const float*)d_in[6];
    const float* gamma2 = (const float*)d_in[7];
    const float* beta2  = (const float*)d_in[8];
    const float* W2     = (const float*)d_in[9];
    const float* b2     = (const float*)d_in[10];
    int Nn   = in_sizes[0] / 64;
    int Ecnt = in_sizes[1];
    float* ws = (float*)d_ws;

    // workspace layout (floats)
    size_t off_nbr  = 0;
    size_t off_cnt  = (size_t)Nn * 64;
    size_t off_deg  = off_cnt + Nn;
    size_t off_sum1 = off_deg + Nn;
    size_t off_ss1  = off_sum1 + 128;
    size_t off_sum2 = off_ss1 + 128;
    size_t off_ss2  = off_sum2 + 64;
    size_t off_w1e  = off_ss2 + 64;
    size_t off_b1e  = off_w1e + 128 * 64;
    size_t off_w2e  = off_b1e + 64;
    size_t off_b2e  = off_w2e + 64 * 64;
    long long zeroN = (long long)(off_sum1 + 384);   // nbr + counters + all stat sums

    float invE = 1.0f / (float)Ecnt;
    int nTiles = (Ecnt + 15) / 16;

    k_zero<<<2048, 256, 0, stream>>>(ws, zeroN);

    long long scatterThreads = 4LL * Ecnt;
    int scatterBlocks = (int)((scatterThreads + 255) / 256);
    k_scatter<<<scatterBlocks, 256, 0, stream>>>(x, src, dst, ws + off_nbr,
                                                 ws + off_cnt, ws + off_deg, Ecnt);

    k_bn1_stats<<<512, 256, 0, stream>>>(x, ws + off_nbr, ws + off_cnt, ws + off_deg,
                                         ws + off_sum1, ws + off_ss1, Nn);

    k_fold1<<<1, 128, 0, stream>>>(ws + off_sum1, ws + off_ss1, gamma1, beta1, W1, b1,
                                   ws + off_w1e, ws + off_b1e, invE);

    k_gemm1_stats<<<2048, 128, 0, stream>>>(x, src, dst, ws + off_nbr, ws + off_w1e,
                                            ws + off_b1e, ws + off_sum2, ws + off_ss2,
                                            nTiles, Ecnt);

    k_fold2<<<1, 64, 0, stream>>>(ws + off_sum2, ws + off_ss2, gamma2, beta2, W2, b2,
                                  ws + off_w2e, ws + off_b2e, invE);

    k_gemm2_out<<<2048, 128, 0, stream>>>(x, src, dst, ws + off_nbr, ws + off_w1e,
                                          ws + off_b1e, ws + off_w2e, ws + off_b2e,
                                          (float*)d_out, nTiles, Ecnt);
}